// SelfAttention_18837726560935
// MI455X (gfx1250) — compile-verified
//
#include <hip/hip_runtime.h>
#include <math.h>

typedef float v2f __attribute__((ext_vector_type(2)));
typedef float v8f __attribute__((ext_vector_type(8)));
typedef unsigned int u32x4 __attribute__((ext_vector_type(4)));
typedef int i32x4 __attribute__((ext_vector_type(4)));
typedef int i32x8 __attribute__((ext_vector_type(8)));

#define BH   128
#define LL   512
#define DD   128
#define LQR  64
#define ROWS 64          // query rows per block
#define NT   128         // threads per block (4 wave32)
#define SQ_ST 132        // padded row stride for sQ/sKV (floats): 132%64=4 -> conflict-free
#define SS_ST 516        // padded row stride for sS: 516%64=4 -> conflict-free
#define TILE_F (16 * SQ_ST)

#if __has_builtin(__builtin_amdgcn_tensor_load_to_lds) && \
    __has_builtin(__builtin_amdgcn_s_wait_tensorcnt)
#define HAVE_TDM 1
#else
#define HAVE_TDM 0
#endif

// D = A(16x4 f32) * B(4x16 f32) + C(16x16 f32), full fp32 matrix op
__device__ __forceinline__ v8f wmma_f32(v2f a, v2f b, v8f c) {
  return __builtin_amdgcn_wmma_f32_16x16x4_f32(
      /*neg_a=*/false, a, /*neg_b=*/false, b,
      /*c_mod=*/(short)0, c, /*reuse_a=*/false, /*reuse_b=*/false);
}

// TDM: DMA a 16-row x 128-float tile (row stride 128 floats in memory) into LDS,
// inserting 4 pad dwords after every 128 dwords so LDS rows land at stride 132.
__device__ __forceinline__ void tdm_load_16x128(const float* gsrc, const float* ldst) {
#if HAVE_TDM
  unsigned long long ga = (unsigned long long)(size_t)gsrc;
  unsigned lds = (unsigned)(size_t)ldst;   // low 32 bits of generic shared ptr = LDS byte offset
  u32x4 g0;
  g0[0] = 1u;                                              // count=1 valid descriptor
  g0[1] = lds;                                             // lds_addr [63:32]
  g0[2] = (unsigned)ga;                                    // global_addr lo
  g0[3] = (unsigned)((ga >> 32) & 0x01ffffffull) | (2u << 30); // global_addr hi | type=2
  i32x8 g1;
  g1[0] = (2 << 16)      // data_size = 4 bytes
        | (1 << 20)      // pad_enable
        | (6 << 22)      // pad_interval: 128 dwords
        | (3 << 25);     // pad_amount: 4 dwords
  g1[1] = (int)(128u << 16);   // tensor_dim0 = 128 (lo16)
  g1[2] = (int)(512u << 16);   // tensor_dim0 hi=0 | tensor_dim1 = 512 (lo16)
  g1[3] = (int)(128u << 16);   // tensor_dim1 hi=0 | tile_dim0 = 128
  g1[4] = 16;                  // tile_dim1 = 16, tile_dim2 = 0
  g1[5] = 128;                 // tensor_dim0_stride = 128 (lo32)
  g1[6] = 0;                   // stride hi | tensor_dim1_stride lo
  g1[7] = 0;
  i32x4 z4 = {0, 0, 0, 0};
  i32x8 z8 = {0, 0, 0, 0, 0, 0, 0, 0};
  // 6-arg form on this toolchain: (g0, g1, g2, g3, g4, cpol)
  __builtin_amdgcn_tensor_load_to_lds(g0, g1, z4, z4, z8, 0);
#else
  (void)gsrc; (void)ldst;
#endif
}

__global__ __launch_bounds__(NT)
void attn_rel_softmax_pv_kernel(const float* __restrict__ Q,
                                const float* __restrict__ K,
                                const float* __restrict__ V,
                                const float* __restrict__ Rel,
                                const float* __restrict__ Mask,
                                float* __restrict__ Out)
{
  __shared__ float sQ[ROWS * SQ_ST];      // 33 KB   : Q row stripe
  __shared__ float sKV[2 * TILE_F];       // 16.5 KB : double-buffered K/V tile
  __shared__ float sS[ROWS * SS_ST];      // 129 KB  : score / prob stripe
  __shared__ float lmask[LL];
  __shared__ float red[NT];
  __shared__ float rowmax[ROWS];
  __shared__ float rowinv[ROWS];

  const int tid  = threadIdx.x;
  const int lane = tid & 31;
  const int wv   = tid >> 5;            // wave id 0..3 -> rows 16*wv..16*wv+15
  const int nidx = lane & 15;
  const int koff = (lane < 16) ? 0 : 2; // fp32 A/B fragment k-offset per ISA layout
  const int b  = blockIdx.y;
  const int qb = blockIdx.x;
  const int q0 = qb * ROWS;

  // ---- stage Q rows (coalesced float4) ----
  for (int i = tid; i < ROWS * DD / 4; i += NT) {
    int r = i >> 5, c4 = i & 31;
    *(float4*)&sQ[r * SQ_ST + 4 * c4] =
        *(const float4*)(Q + ((size_t)b * LL + q0 + r) * DD + 4 * c4);
  }

  // ---- phase 1: S = Q * K^T  (fp32 WMMA, K tiles streamed by TDM, double-buffered) ----
#if HAVE_TDM
  if (wv == 0) tdm_load_16x128(K + (size_t)b * LL * DD, &sKV[0]);
#endif
  for (int j = 0; j < LL / 16; ++j) {
#if HAVE_TDM
    if (wv == 0) __builtin_amdgcn_s_wait_tensorcnt((short)0);  // tile j landed
    __syncthreads();                                           // publish; frees buf (j+1)&1
    if (wv == 0 && (j + 1) < LL / 16)
      tdm_load_16x128(K + ((size_t)b * LL + 16 * (j + 1)) * DD,
                      &sKV[((j + 1) & 1) * TILE_F]);
    const int kb = (j & 1) * TILE_F;
#else
    __syncthreads();
    for (int i = tid; i < 16 * DD / 4; i += NT) {
      int r = i >> 5, c4 = i & 31;
      *(float4*)&sKV[r * SQ_ST + 4 * c4] =
          *(const float4*)(K + ((size_t)b * LL + 16 * j + r) * DD + 4 * c4);
    }
    __syncthreads();
    const int kb = 0;
#endif
    v8f c = {};
    const float* aq = &sQ[(16 * wv + nidx) * SQ_ST + koff];
    const float* bk = &sKV[kb + nidx * SQ_ST + koff];
    #pragma unroll 8
    for (int d = 0; d < DD; d += 4) {
      v2f a  = *(const v2f*)(aq + d);
      v2f bb = *(const v2f*)(bk + d);
      c = wmma_f32(a, bb, c);
    }
    #pragma unroll
    for (int r = 0; r < 8; ++r) {
      int m = (lane < 16) ? r : (r + 8);
      sS[(16 * wv + m) * SS_ST + 16 * j + nidx] = c[r];
    }
  }
#if HAVE_TDM
  // kick off V tile 0 now; softmax below hides its latency.
  // (buffer 0 was last read at iteration j=30; the j=31 barrier retired those reads)
  if (wv == 0) tdm_load_16x128(V + (size_t)b * LL * DD, &sKV[0]);
#endif
  __syncthreads();

  // ---- mask log term ----
  for (int i = tid; i < LL; i += NT)
    lmask[i] = logf(1.0f - Mask[(size_t)b * LL + i]);

  // ---- relative-position term (last 64 rows x last 64 cols only) ----
  if (qb == (LL / ROWS - 1)) {
    for (int p = tid; p < LQR * LQR; p += NT) {
      int q = p >> 6, kk = p & 63;     // local q == rel q index here (q0 == L-LQ)
      const float* rp = Rel + (((size_t)b * LQR + q) * LQR + kk) * DD;
      if (p + NT < LQR * LQR) {        // prefetch next pair's row (global_prefetch_b8)
        int q2 = (p + NT) >> 6, k2 = (p + NT) & 63;
        __builtin_prefetch(Rel + (((size_t)b * LQR + q2) * LQR + k2) * DD, 0, 1);
      }
      const float* qp = &sQ[q * SQ_ST];
      float acc = 0.f;
      for (int d = 0; d < DD; d += 4) {
        float4 rv = *(const float4*)(rp + d);
        acc += rv.x * qp[d] + rv.y * qp[d + 1] + rv.z * qp[d + 2] + rv.w * qp[d + 3];
      }
      sS[q * SS_ST + (LL - LQR) + kk] += acc;
    }
  }
  __syncthreads();

  // ---- softmax over each row of the 64x512 stripe ----
  const float scale = 0.08838834764831845f;  // 1/sqrt(128)
  {
    int r = tid >> 1, h = tid & 1;           // 2 threads per row
    const float* srow = &sS[r * SS_ST];
    float mx = -3.4e38f;
    for (int c0 = h * 256; c0 < h * 256 + 256; ++c0)
      mx = fmaxf(mx, srow[c0] * scale + lmask[c0]);
    red[tid] = mx;
    __syncthreads();
    if (tid < ROWS) rowmax[tid] = fmaxf(red[2 * tid], red[2 * tid + 1]);
    __syncthreads();
    float rm = rowmax[r];
    float sm = 0.f;
    for (int c0 = h * 256; c0 < h * 256 + 256; ++c0)
      sm += expf(srow[c0] * scale + lmask[c0] - rm);
    red[tid] = sm;
    __syncthreads();
    if (tid < ROWS) rowinv[tid] = 1.0f / (red[2 * tid] + red[2 * tid + 1]);
    __syncthreads();
  }

  // ---- normalize, write attn output, keep P in LDS ----
  float* attn_out = Out + (size_t)BH * LL * DD;   // d_out = out ‖ attn
  for (int i = tid; i < ROWS * LL; i += NT) {
    int r = i >> 9, c = i & (LL - 1);
    float v = expf(sS[r * SS_ST + c] * scale + lmask[c] - rowmax[r]) * rowinv[r];
    sS[r * SS_ST + c] = v;
    attn_out[((size_t)b * LL + q0 + r) * LL + c] = v;
  }

  // ---- phase 2: out = P * V (V tiles streamed by TDM, double-buffered) ----
  v8f acc[8] = {};
  for (int t = 0; t < LL / 16; ++t) {
#if HAVE_TDM
    if (wv == 0) __builtin_amdgcn_s_wait_tensorcnt((short)0);  // tile t landed
    __syncthreads();
    if (wv == 0 && (t + 1) < LL / 16)
      tdm_load_16x128(V + ((size_t)b * LL + 16 * (t + 1)) * DD,
                      &sKV[((t + 1) & 1) * TILE_F]);
    const int vb = (t & 1) * TILE_F;
#else
    __syncthreads();
    for (int i = tid; i < 16 * DD / 4; i += NT) {
      int r = i >> 5, c4 = i & 31;
      *(float4*)&sKV[r * SQ_ST + 4 * c4] =
          *(const float4*)(V + ((size_t)b * LL + 16 * t + r) * DD + 4 * c4);
    }
    __syncthreads();
    const int vb = 0;
#endif
    const int k0 = 16 * t;
    #pragma unroll
    for (int kk = 0; kk < 16; kk += 4) {
      v2f a = *(const v2f*)&sS[(16 * wv + nidx) * SS_ST + k0 + kk + koff];
      #pragma unroll
      for (int dt = 0; dt < 8; ++dt) {
        v2f bb;
        bb.x = sKV[vb + (kk + koff) * SQ_ST + dt * 16 + nidx];
        bb.y = sKV[vb + (kk + koff + 1) * SQ_ST + dt * 16 + nidx];
        acc[dt] = wmma_f32(a, bb, acc[dt]);
      }
    }
  }

  #pragma unroll
  for (int dt = 0; dt < 8; ++dt) {
    #pragma unroll
    for (int r = 0; r < 8; ++r) {
      int m = (lane < 16) ? r : (r + 8);
      Out[((size_t)b * LL + q0 + 16 * wv + m) * DD + dt * 16 + nidx] = acc[dt][r];
    }
  }
}

extern "C" void kernel_launch(void* const* d_in, const int* in_sizes, int n_in,
                              void* d_out, int out_size, void* d_ws, size_t ws_size,
                              hipStream_t stream) {
  (void)in_sizes; (void)n_in; (void)out_size; (void)d_ws; (void)ws_size;
  const float* Q    = (const float*)d_in[0];
  const float* K    = (const float*)d_in[1];
  const float* V    = (const float*)d_in[2];
  const float* Rel  = (const float*)d_in[3];
  const float* Mask = (const float*)d_in[4];
  float* Out = (float*)d_out;

  dim3 grid(LL / ROWS, BH);   // (8, 128)
  dim3 block(NT);             // 128 threads = 4 wave32
  attn_rel_softmax_pv_kernel<<<grid, block, 0, stream>>>(Q, K, V, Rel, Mask, Out);
}